// Attn_2946347565574
// MI455X (gfx1250) — compile-verified
//
#include <hip/hip_runtime.h>
#include <math.h>

#define B_    64
#define S_    2048
#define D_    1024
#define SPLIT 8
#define CHUNK (S_ / SPLIT)   // 256 rows per chunk

typedef __attribute__((ext_vector_type(2))) float v2f;
typedef __attribute__((ext_vector_type(8))) float v8f;

// workspace layout (float offsets)
#define WS_G    0                         // B*D          : G = H @ W
#define WS_C    (B_ * D_)                 // B            : c[b] = h.bias
#define WS_M    (WS_C + B_)               // B*SPLIT      : chunk max
#define WS_L    (WS_M + B_ * SPLIT)       // B*SPLIT      : chunk sum
#define WS_CTX  (WS_L + B_ * SPLIT)       // B*SPLIT*D    : chunk context partials

// ---------------------------------------------------------------------------
// Kernel 1: G = H @ W  via V_WMMA_F32_16X16X4_F32 (exact fp32 matrix path)
// One wave per 16x16 tile. A: 16x4 (lanes 0-15: M rows, K=0,1 in v0,v1;
// lanes 16-31: K=2,3). B: 4x16 mirrored. C/D: vgpr r = row r + half*8.
// ---------------------------------------------------------------------------
__global__ __launch_bounds__(32)
void gemm_hW(const float* __restrict__ H, const float* __restrict__ W,
             float* __restrict__ G) {
    const int lane  = threadIdx.x;
    const int half  = lane >> 4;     // 0 or 1
    const int l16   = lane & 15;
    const int mbase = blockIdx.x * 16;
    const int nbase = blockIdx.y * 16;

    const float* arow = H + (size_t)(mbase + l16) * D_;

    v8f c = {};
    for (int k0 = 0; k0 < D_; k0 += 4) {
        const int ka = k0 + half * 2;
        v2f a, b;
        a.x = arow[ka];
        a.y = arow[ka + 1];
        b.x = W[(size_t)ka * D_ + nbase + l16];
        b.y = W[(size_t)(ka + 1) * D_ + nbase + l16];
        // 8 args: (neg_a, A, neg_b, B, c_mod, C, reuse_a, reuse_b)
        c = __builtin_amdgcn_wmma_f32_16x16x4_f32(
                false, a, false, b, (short)0, c, false, false);
    }
#pragma unroll
    for (int r = 0; r < 8; ++r) {
        G[(size_t)(mbase + r + half * 8) * D_ + nbase + l16] = c[r];
    }
}

// ---------------------------------------------------------------------------
// Kernel 2: c[b] = h[b] . bias   (one wave per batch)
// ---------------------------------------------------------------------------
__global__ __launch_bounds__(32)
void hdotb(const float* __restrict__ H, const float* __restrict__ bias,
           float* __restrict__ Cb) {
    const int b    = blockIdx.x;
    const int lane = threadIdx.x;
    const float* hr = H + (size_t)b * D_;
    float acc = 0.f;
    for (int j = lane; j < D_; j += 32) acc += hr[j] * bias[j];
    for (int off = 16; off > 0; off >>= 1) acc += __shfl_xor(acc, off, 32);
    if (lane == 0) Cb[b] = acc;
}

// ---------------------------------------------------------------------------
// Kernel 3: per-(batch, chunk) energies + partial softmax + partial context.
// Block = 256 threads (8 wave32). Chunk = 256 source rows.
// ---------------------------------------------------------------------------
__global__ __launch_bounds__(256)
void attn_partial(const float* __restrict__ enc, const float* __restrict__ mask,
                  const float* __restrict__ G, const float* __restrict__ Cb,
                  float* __restrict__ m_part, float* __restrict__ l_part,
                  float* __restrict__ ctx_part) {
    __shared__ float g_s[D_];
    __shared__ float e_s[CHUNK];
    __shared__ float red_s[8];
    __shared__ float bcast;

    const int tid   = threadIdx.x;
    const int wave  = tid >> 5;
    const int lane  = tid & 31;
    const int b     = blockIdx.x / SPLIT;
    const int chunk = blockIdx.x % SPLIT;
    const int s0    = chunk * CHUNK;

    // stage G row for this batch into LDS (4 KB)
    for (int j = tid; j < D_; j += 256) g_s[j] = G[(size_t)b * D_ + j];
    const float cb = Cb[b];
    __syncthreads();

    // ---- Phase A: energies, one row per wave iteration --------------------
    for (int sl = wave; sl < CHUNK; sl += 8) {
        const float* row = enc + ((size_t)b * S_ + s0 + sl) * D_;
        float acc = 0.f;
#pragma unroll 8
        for (int j = 0; j < D_ / 32; ++j) {
            const int idx = lane + 32 * j;          // coalesced 128B per wave
            acc += row[idx] * g_s[idx];
        }
        for (int off = 16; off > 0; off >>= 1) acc += __shfl_xor(acc, off, 32);
        if (lane == 0) {
            float e = mask[(size_t)b * S_ + s0 + sl] * (acc + cb);
            if (e == 0.0f) e = -1.0e10f;            // match reference semantics
            e_s[sl] = e;
        }
    }
    __syncthreads();

    // ---- Phase B1: chunk max ---------------------------------------------
    const float v = e_s[tid];
    float mval = v;
    for (int off = 16; off > 0; off >>= 1)
        mval = fmaxf(mval, __shfl_xor(mval, off, 32));
    if (lane == 0) red_s[wave] = mval;
    __syncthreads();
    if (tid == 0) {
        float m = red_s[0];
        for (int i = 1; i < 8; ++i) m = fmaxf(m, red_s[i]);
        bcast = m;
    }
    __syncthreads();
    const float m = bcast;

    // ---- Phase B2: p = exp(e-m), chunk sum --------------------------------
    const float p = __expf(v - m);
    e_s[tid] = p;                                   // reuse LDS for weights
    float lsum = p;
    for (int off = 16; off > 0; off >>= 1) lsum += __shfl_xor(lsum, off, 32);
    if (lane == 0) red_s[wave] = lsum;
    __syncthreads();
    if (tid == 0) {
        float l = 0.f;
        for (int i = 0; i < 8; ++i) l += red_s[i];
        m_part[blockIdx.x] = m;
        l_part[blockIdx.x] = l;
    }

    // ---- Phase B3: partial context (unnormalized) -------------------------
    // thread t owns columns t, t+256, t+512, t+768 -> coalesced loads,
    // chunk data is hot in L2 from phase A.
    float a0 = 0.f, a1 = 0.f, a2 = 0.f, a3 = 0.f;
    const float* encb = enc + ((size_t)b * S_ + s0) * D_;
    for (int s = 0; s < CHUNK; ++s) {
        const float w = e_s[s];                     // LDS broadcast
        const float* r = encb + (size_t)s * D_;
        a0 += w * r[tid];
        a1 += w * r[tid + 256];
        a2 += w * r[tid + 512];
        a3 += w * r[tid + 768];
    }
    float* cp = ctx_part + (size_t)blockIdx.x * D_;
    cp[tid]       = a0;
    cp[tid + 256] = a1;
    cp[tid + 512] = a2;
    cp[tid + 768] = a3;
}

// ---------------------------------------------------------------------------
// Kernel 4: combine SPLIT partials per batch, normalize.
// ---------------------------------------------------------------------------
__global__ __launch_bounds__(256)
void attn_reduce(const float* __restrict__ m_part, const float* __restrict__ l_part,
                 const float* __restrict__ ctx_part, float* __restrict__ out) {
    const int b   = blockIdx.x;
    const int tid = threadIdx.x;

    float mv[SPLIT];
    float M = -INFINITY;
#pragma unroll
    for (int i = 0; i < SPLIT; ++i) {
        mv[i] = m_part[b * SPLIT + i];
        M = fmaxf(M, mv[i]);
    }
    float sc[SPLIT];
    float L = 0.f;
#pragma unroll
    for (int i = 0; i < SPLIT; ++i) {
        sc[i] = __expf(mv[i] - M);
        L += l_part[b * SPLIT + i] * sc[i];
    }
    const float invL = 1.0f / L;

#pragma unroll
    for (int j = 0; j < 4; ++j) {
        const int d = tid + 256 * j;
        float s = 0.f;
#pragma unroll
        for (int i = 0; i < SPLIT; ++i)
            s += sc[i] * ctx_part[(size_t)(b * SPLIT + i) * D_ + d];
        out[(size_t)b * D_ + d] = s * invL;
    }
}

// ---------------------------------------------------------------------------
extern "C" void kernel_launch(void* const* d_in, const int* in_sizes, int n_in,
                              void* d_out, int out_size, void* d_ws, size_t ws_size,
                              hipStream_t stream) {
    const float* hidden = (const float*)d_in[0];   // [1,B,D]
    const float* enc    = (const float*)d_in[1];   // [B,S,D]
    const float* mask   = (const float*)d_in[2];   // [B,S]
    const float* W      = (const float*)d_in[3];   // [D,D]
    const float* bias   = (const float*)d_in[4];   // [D]
    float* out = (float*)d_out;                    // [B,D]

    float* ws = (float*)d_ws;
    float* G  = ws + WS_G;
    float* Cb = ws + WS_C;
    float* Mp = ws + WS_M;
    float* Lp = ws + WS_L;
    float* Cx = ws + WS_CTX;

    gemm_hW<<<dim3(B_ / 16, D_ / 16), 32, 0, stream>>>(hidden, W, G);
    hdotb<<<B_, 32, 0, stream>>>(hidden, bias, Cb);
    attn_partial<<<B_ * SPLIT, 256, 0, stream>>>(enc, mask, G, Cb, Mp, Lp, Cx);
    attn_reduce<<<B_, 256, 0, stream>>>(Mp, Lp, Cx, out);
}